// FractalEmbedding_30365418782764
// MI455X (gfx1250) — compile-verified
//
#include <hip/hip_runtime.h>

// FractalEmbedding on gfx1250:
//   feats[m,k]: 8-step Julia iteration per token (k = interleaved zr/zi)
//   out[m,n]  = sum_k feats[m,k] * W[n,k] * scale
// M=32768, K=16, N=2048 -> store-bandwidth-bound (268 MB fp32 out, ~11.5us
// floor at 23.3 TB/s). GEMM on the f32 matrix pipe: 4 chained
// V_WMMA_F32_16X16X4_F32 per 16x16 output tile, fp32 end-to-end.

typedef __attribute__((ext_vector_type(2))) float v2f;
typedef __attribute__((ext_vector_type(8))) float v8f;

#define STEPS 8
#define KDIM 16          // 2*STEPS
#define DDIM 2048        // N
#define WAVES_PER_BLOCK 8
#define NTILES (DDIM / 16)                       // 128
#define TILES_PER_WAVE (NTILES / WAVES_PER_BLOCK) // 16

__global__ __launch_bounds__(256)
void fractal_embed_wmma(const float* __restrict__ cs,
                        const float* __restrict__ W,
                        const float* __restrict__ scale_p,
                        float* __restrict__ out)
{
    const int  lane = threadIdx.x & 31;
    const int  l16  = lane & 15;
    const bool hi   = (lane >> 4) != 0;  // lanes 16-31 carry K+2/K+3 of each block

    // Wave id is uniform across the wave: pin it to an SGPR so the tile loop
    // runs on scalar control flow (s_cmp/s_cbranch) instead of exec masking.
    const int wave_s = __builtin_amdgcn_readfirstlane((int)(threadIdx.x >> 5));

    const int m0   = blockIdx.x * 16;   // 16 token rows per block
    const int mrow = m0 + l16;          // token this lane iterates

    // ---- Julia iteration z <- z^2 + c from z=0. A-fragment f32 16x4 layout:
    //   lanes 0-15 : VGPR0 = K=kb,   VGPR1 = K=kb+1   (row = lane)
    //   lanes 16-31: VGPR0 = K=kb+2, VGPR1 = K=kb+3   (row = lane-16)
    // Fragment j covers K = 4j..4j+3, i.e. step 2j (lo half) / 2j+1 (hi half).
    // Build fragments inline: one compare, 8 cndmasks, no register array.
    const float cr = cs[2 * mrow + 0];
    const float ci = cs[2 * mrow + 1];
    v2f a[4];
    {
        float zr = 0.0f, zi = 0.0f;
#pragma unroll
        for (int j = 0; j < 4; ++j) {
            // step 2j
            const float er = zr * zr - zi * zi + cr;
            const float ei = 2.0f * zr * zi + ci;
            // step 2j+1
            const float or_ = er * er - ei * ei + cr;
            const float oi  = 2.0f * er * ei + ci;
            a[j].x = hi ? or_ : er;
            a[j].y = hi ? oi  : ei;
            zr = or_; zi = oi;
        }
    }

    const float scale = scale_p[0];

    // ---- Each wave walks exactly TILES_PER_WAVE N-tiles of 16 columns;
    // A fragments are reused across all of them, W rows stay L2-hot.
    const int   col0  = wave_s * 16 + l16;
    const float* wrow = W + (size_t)col0 * KDIM + (hi ? 2 : 0);
    float*       o    = out + (size_t)(m0 + (hi ? 8 : 0)) * DDIM + col0;

    for (int it = 0; it < TILES_PER_WAVE; ++it) {
        v8f acc = {};
#pragma unroll
        for (int j = 0; j < 4; ++j) {
            const v2f b = *(const v2f*)(wrow + 4 * j);   // 8B-aligned b64 load
            acc = __builtin_amdgcn_wmma_f32_16x16x4_f32(
                /*neg_a=*/false, a[j],
                /*neg_b=*/false, b,
                /*c_mod=*/(short)0, acc,
                /*reuse_a=*/false, /*reuse_b=*/false);
        }

        // Store 16x16 tile. C/D layout: VGPR v -> row m0 + v + 8*hi, col0.
        // 268 MB streamed write-once output (> 192 MB L2) -> non-temporal
        // stores keep W/cs resident in L2.
#pragma unroll
        for (int v = 0; v < 8; ++v) {
            __builtin_nontemporal_store(acc[v] * scale, o + (size_t)v * DDIM);
        }

        // Advance 8 tiles (WAVES_PER_BLOCK stride) = 128 columns.
        wrow += (size_t)WAVES_PER_BLOCK * 16 * KDIM;
        o    += (size_t)WAVES_PER_BLOCK * 16;
    }
}

extern "C" void kernel_launch(void* const* d_in, const int* in_sizes, int n_in,
                              void* d_out, int out_size, void* d_ws, size_t ws_size,
                              hipStream_t stream) {
    // setup_inputs order: token_ids (unused), cs [B,L,2] f32, W [D,16] f32,
    // scale [] f32. Output: [B,L,D] f32.
    const float* cs      = (const float*)d_in[1];
    const float* W       = (const float*)d_in[2];
    const float* scale_p = (const float*)d_in[3];
    float*       out     = (float*)d_out;

    const int M = in_sizes[1] / 2;          // B*L = 32768
    dim3 grid(M / 16);                      // 2048 blocks, one 16-row M tile each
    dim3 block(256);                        // 8 wave32s stride the 128 N tiles
    hipLaunchKernelGGL(fractal_embed_wmma, grid, block, 0, stream,
                       cs, W, scale_p, out);
}